// SteerRNN_71622874628691
// MI455X (gfx1250) — compile-verified
//
#include <hip/hip_runtime.h>
#include <cmath>

typedef __attribute__((ext_vector_type(2))) float v2f;
typedef __attribute__((ext_vector_type(8))) float v8f;

#define B_    32
#define T_    2048
#define DM    1024
#define DMLP  128
#define DH    64
#define DZ    64
#define M_TOTAL (B_ * T_)

// Async global->LDS copy (GLOBAL_LOAD_ASYNC_TO_LDS_B128, tracked by ASYNCcnt).
#define USE_ASYNC_LDS 1

// V_WMMA_F32_16X16X4_F32: D(16x16 f32) = A(16x4 f32) * B(4x16 f32) + C
__device__ __forceinline__ v8f wmma_f32_k4(v2f a, v2f b, v8f c) {
  return __builtin_amdgcn_wmma_f32_16x16x4_f32(false, a, false, b, (short)0, c,
                                               false, false);
}

__device__ __forceinline__ float gelu_exact(float x) {
  return 0.5f * x * (1.0f + erff(x * 0.7071067811865476f));
}

__device__ __forceinline__ float sigmoidf_(float x) {
  return 1.0f / (1.0f + expf(-x));
}

// ---------------------------------------------------------------------------
// Kernel 1: pre = r_src @ lat_w1[:1024,:] + lat_b1     (M=65536, N=128, K=1024)
// block = 128 threads (4 waves); each wave owns 16 M-rows x 128 N cols.
// A-frag layout (16x4 f32): lanes 0-15 hold {K=k,k+1}, lanes 16-31 {k+2,k+3}.
// ---------------------------------------------------------------------------
__global__ void __launch_bounds__(128) pre_kernel(const float* __restrict__ r,
                                                  const float* __restrict__ w1,
                                                  const float* __restrict__ b1,
                                                  float* __restrict__ pre) {
  const int wave = threadIdx.x >> 5;
  const int lane = threadIdx.x & 31;
  const int half = lane >> 4;   // 0: K=k,k+1   1: K=k+2,k+3
  const int nl   = lane & 15;

  const int row = blockIdx.x * 64 + wave * 16 + nl;  // A row this lane loads
  const float* arow = r + (size_t)row * DM;

  v8f acc[8] = {};
  for (int k = 0; k < DM; k += 4) {
    if ((k & 127) == 0) __builtin_prefetch(arow + k + 128, 0, 1);
    v2f a;
    a.x = arow[k + 2 * half];
    a.y = arow[k + 2 * half + 1];
#pragma unroll
    for (int n = 0; n < 8; ++n) {
      const int col = n * 16 + nl;
      v2f b;
      b.x = w1[(size_t)(k + 2 * half)     * DMLP + col];
      b.y = w1[(size_t)(k + 2 * half + 1) * DMLP + col];
      acc[n] = wmma_f32_k4(a, b, acc[n]);
    }
  }
  // C/D layout: lanes<16 -> M=v, lanes>=16 -> M=v+8; N = lane&15
  const int mbase = blockIdx.x * 64 + wave * 16 + 8 * half;
#pragma unroll
  for (int n = 0; n < 8; ++n) {
    const int col = n * 16 + nl;
    const float bias = b1[col];
#pragma unroll
    for (int v = 0; v < 8; ++v)
      pre[(size_t)(mbase + v) * DMLP + col] = acc[n][v] + bias;
  }
}

// ---------------------------------------------------------------------------
// Kernel 2: the sequential scan. One workgroup (256 threads); h, hid, z live
// in LDS; only the K=64 recurrent matmuls + softmax + GRU gates are serial.
// Writes feat[b,t,:] = [z(64), h_new(64)] for the parallel head kernels.
// ---------------------------------------------------------------------------
__global__ void __launch_bounds__(256) scan_kernel(
    const float* __restrict__ pre,                         // (B*T,128), b1 included
    const float* __restrict__ w1h,                         // lat_w1[1024:1088,:]
    const float* __restrict__ w2, const float* __restrict__ b2,   // (128,64)
    const float* __restrict__ wi, const float* __restrict__ bi,   // (64,192)
    const float* __restrict__ wh, const float* __restrict__ bh,   // (64,192)
    float* __restrict__ feat) {                            // (B*T,128)
  __shared__ float sh[2][B_ * DH];     // double-buffered hidden state
  __shared__ float shid[B_ * DMLP];
  __shared__ float sz[B_ * DZ];
  const int tid = threadIdx.x;

  for (int i = tid; i < B_ * DH; i += 256) sh[0][i] = 0.0f;
  __syncthreads();

  int cur = 0;
  for (int t = 0; t < T_; ++t) {
    const float* hc = sh[cur];

    // stage 1: hid = gelu(pre[:,t,:] + h @ w1h)   (32x128, K=64)
    {
      const int j  = tid & 127;
      const int bg = (tid >> 7) * 16;
      float acc[16];
#pragma unroll
      for (int r0 = 0; r0 < 16; ++r0)
        acc[r0] = pre[((size_t)(bg + r0) * T_ + t) * DMLP + j];
      for (int k = 0; k < DH; ++k) {
        const float w = w1h[k * DMLP + j];
#pragma unroll
        for (int r0 = 0; r0 < 16; ++r0) acc[r0] += hc[(bg + r0) * DH + k] * w;
      }
#pragma unroll
      for (int r0 = 0; r0 < 16; ++r0)
        shid[(bg + r0) * DMLP + j] = gelu_exact(acc[r0]);
    }
    __syncthreads();

    // stage 2: logits = hid @ w2 + b2; softmax over 8-class groups -> z
    {
      const int b   = tid >> 3;
      const int cat = tid & 7;
      float l[8];
#pragma unroll
      for (int c = 0; c < 8; ++c) l[c] = b2[cat * 8 + c];
      for (int k = 0; k < DMLP; ++k) {
        const float hv = shid[b * DMLP + k];
#pragma unroll
        for (int c = 0; c < 8; ++c) l[c] += hv * w2[k * DZ + cat * 8 + c];
      }
      float mx = l[0];
#pragma unroll
      for (int c = 1; c < 8; ++c) mx = fmaxf(mx, l[c]);
      float s = 0.0f;
#pragma unroll
      for (int c = 0; c < 8; ++c) { l[c] = expf(l[c] - mx); s += l[c]; }
      const float inv = 1.0f / s;
#pragma unroll
      for (int c = 0; c < 8; ++c) sz[b * DZ + cat * 8 + c] = l[c] * inv;
    }
    __syncthreads();

    // stage 3: GRU gates (torch order r,z,n), h_new, emit feat
    {
      const int j  = tid & 63;
      const int bg = (tid >> 6) * 8;
      float ir[8], iz[8], inn[8], hr[8], hz[8], hn[8];
#pragma unroll
      for (int r0 = 0; r0 < 8; ++r0) {
        ir[r0] = bi[j];  iz[r0] = bi[DH + j];  inn[r0] = bi[2 * DH + j];
        hr[r0] = bh[j];  hz[r0] = bh[DH + j];  hn[r0]  = bh[2 * DH + j];
      }
      for (int k = 0; k < DZ; ++k) {
        const float wir = wi[k * 3 * DH + j];
        const float wiz = wi[k * 3 * DH + DH + j];
        const float win = wi[k * 3 * DH + 2 * DH + j];
        const float whr = wh[k * 3 * DH + j];
        const float whz = wh[k * 3 * DH + DH + j];
        const float whn = wh[k * 3 * DH + 2 * DH + j];
#pragma unroll
        for (int r0 = 0; r0 < 8; ++r0) {
          const float zv = sz[(bg + r0) * DZ + k];
          const float hv = hc[(bg + r0) * DH + k];
          ir[r0] += zv * wir;  iz[r0] += zv * wiz;  inn[r0] += zv * win;
          hr[r0] += hv * whr;  hz[r0] += hv * whz;  hn[r0]  += hv * whn;
        }
      }
#pragma unroll
      for (int r0 = 0; r0 < 8; ++r0) {
        const int b = bg + r0;
        const float rg   = sigmoidf_(ir[r0] + hr[r0]);
        const float u    = sigmoidf_(iz[r0] + hz[r0]);
        const float nn   = tanhf(inn[r0] + rg * hn[r0]);
        const float hnew = (1.0f - u) * nn + u * hc[b * DH + j];
        sh[cur ^ 1][b * DH + j] = hnew;
        float* frow = feat + ((size_t)b * T_ + t) * DMLP;
        frow[DZ + j] = hnew;
        frow[j]      = sz[b * DZ + j];
      }
    }
    __syncthreads();
    cur ^= 1;
  }
}

// ---------------------------------------------------------------------------
// Kernel 3: fused head MLPs.  grid.y selects head (0=steer, 1=rec, 2=loss).
// feat tile staged global->LDS with GLOBAL_LOAD_ASYNC_TO_LDS_B128 (ASYNCcnt),
// then H = gelu(feat @ w1 + b1) via WMMA -> LDS -> out = H @ w2 + b2 via WMMA.
// ---------------------------------------------------------------------------
__global__ void __launch_bounds__(128) head_kernel(
    const float* __restrict__ feat,
    const float* __restrict__ sw1, const float* __restrict__ sb1,
    const float* __restrict__ sw2, const float* __restrict__ sb2,
    const float* __restrict__ rw1, const float* __restrict__ rb1,
    const float* __restrict__ rw2, const float* __restrict__ rb2,
    const float* __restrict__ lw1, const float* __restrict__ lb1,
    const float* __restrict__ lw2, const float* __restrict__ lb2,
    float* __restrict__ out_s, float* __restrict__ out_r,
    float* __restrict__ out_l) {
  __shared__ float sM[64 * DMLP];  // feat tile, then reused for H

  const int head = blockIdx.y;
  const float* w1 = head == 0 ? sw1 : (head == 1 ? rw1 : lw1);
  const float* b1 = head == 0 ? sb1 : (head == 1 ? rb1 : lb1);
  const int m0 = blockIdx.x * 64;

#if USE_ASYNC_LDS
  {  // stage feat tile (64x128 f32 = 32KB contiguous) via async copy to LDS.
     // 128 lanes x 16B x 16 issues.  VDST = per-lane LDS byte address
     // (low 32 bits of the generic LDS pointer = flat-LDS offset).
    const char* gbase = (const char*)(feat + (size_t)m0 * DMLP);
    unsigned lbase = (unsigned)(size_t)(&sM[0]);
#pragma unroll
    for (int i = 0; i < 16; ++i) {
      const unsigned off = (threadIdx.x + i * 128) * 16u;
      unsigned lds_addr = lbase + off;
      unsigned long long gaddr = (unsigned long long)(gbase + off);
      asm volatile("global_load_async_to_lds_b128 %0, %1, off"
                   :
                   : "v"(lds_addr), "v"(gaddr)
                   : "memory");
    }
    asm volatile("s_wait_asynccnt 0x0" ::: "memory");
  }
#else
  {  // fallback: staged through VGPRs
    const float4* src = (const float4*)(feat + (size_t)m0 * DMLP);
    float4* dst = (float4*)sM;
    for (int i = threadIdx.x; i < 64 * DMLP / 4; i += 128) dst[i] = src[i];
  }
#endif
  __syncthreads();

  const int wave = threadIdx.x >> 5;
  const int lane = threadIdx.x & 31;
  const int half = lane >> 4;
  const int nl   = lane & 15;
  const int ar   = wave * 16 + nl;

  // GEMM1: 16x128 tile of H, K=128
  v8f acc[8] = {};
  for (int k = 0; k < DMLP; k += 4) {
    v2f a;
    a.x = sM[ar * DMLP + k + 2 * half];
    a.y = sM[ar * DMLP + k + 2 * half + 1];
#pragma unroll
    for (int n = 0; n < 8; ++n) {
      const int col = n * 16 + nl;
      v2f b;
      b.x = w1[(k + 2 * half)     * DMLP + col];
      b.y = w1[(k + 2 * half + 1) * DMLP + col];
      acc[n] = wmma_f32_k4(a, b, acc[n]);
    }
  }
  __syncthreads();  // everyone finished reading feat from sM

  {  // H = gelu(acc + b1) -> sM  (C layout: M = v + 8*half)
    const int mb = wave * 16 + 8 * half;
#pragma unroll
    for (int n = 0; n < 8; ++n) {
      const int col = n * 16 + nl;
      const float bias = b1[col];
#pragma unroll
      for (int v = 0; v < 8; ++v)
        sM[(mb + v) * DMLP + col] = gelu_exact(acc[n][v] + bias);
    }
  }
  __syncthreads();

  if (head < 2) {
    const float* w2 = head == 0 ? sw2 : rw2;
    const float* b2 = head == 0 ? sb2 : rb2;
    float* outp     = head == 0 ? out_s : out_r;
    for (int nb = 0; nb < DM; nb += 128) {
      v8f a2[8] = {};
      for (int k = 0; k < DMLP; k += 4) {
        v2f a;
        a.x = sM[ar * DMLP + k + 2 * half];
        a.y = sM[ar * DMLP + k + 2 * half + 1];
#pragma unroll
        for (int n = 0; n < 8; ++n) {
          const int col = nb + n * 16 + nl;
          v2f b;
          b.x = w2[(size_t)(k + 2 * half)     * DM + col];
          b.y = w2[(size_t)(k + 2 * half + 1) * DM + col];
          a2[n] = wmma_f32_k4(a, b, a2[n]);
        }
      }
      const int mb = m0 + wave * 16 + 8 * half;
#pragma unroll
      for (int n = 0; n < 8; ++n) {
        const int col = nb + n * 16 + nl;
        const float bias = b2[col];
#pragma unroll
        for (int v = 0; v < 8; ++v)
          outp[(size_t)(mb + v) * DM + col] = a2[n][v] + bias;
      }
    }
  } else {
    // loss head second layer: (64x128) @ (128x1) -> 64 scalars
    if (threadIdx.x < 64) {
      float a = lb2[0];
      for (int k = 0; k < DMLP; ++k) a += sM[threadIdx.x * DMLP + k] * lw2[k];
      out_l[m0 + threadIdx.x] = a;
    }
  }
}

// ---------------------------------------------------------------------------
extern "C" void kernel_launch(void* const* d_in, const int* in_sizes, int n_in,
                              void* d_out, int out_size, void* d_ws,
                              size_t ws_size, hipStream_t stream) {
  const float* r_src   = (const float*)d_in[0];
  const float* lat_w1  = (const float*)d_in[1];
  const float* lat_b1  = (const float*)d_in[2];
  const float* lat_w2  = (const float*)d_in[3];
  const float* lat_b2  = (const float*)d_in[4];
  const float* gru_wi  = (const float*)d_in[5];
  const float* gru_bi  = (const float*)d_in[6];
  const float* gru_wh  = (const float*)d_in[7];
  const float* gru_bh  = (const float*)d_in[8];
  const float* steer_w1 = (const float*)d_in[9];
  const float* steer_b1 = (const float*)d_in[10];
  const float* steer_w2 = (const float*)d_in[11];
  const float* steer_b2 = (const float*)d_in[12];
  const float* rec_w1  = (const float*)d_in[13];
  const float* rec_b1  = (const float*)d_in[14];
  const float* rec_w2  = (const float*)d_in[15];
  const float* rec_b2  = (const float*)d_in[16];
  const float* loss_w1 = (const float*)d_in[17];
  const float* loss_b1 = (const float*)d_in[18];
  const float* loss_w2 = (const float*)d_in[19];
  const float* loss_b2 = (const float*)d_in[20];

  float* ws   = (float*)d_ws;
  float* pre  = ws;                                 // (B*T, 128)
  float* feat = ws + (size_t)M_TOTAL * DMLP;        // (B*T, 128)

  float* out_s = (float*)d_out;                     // (B,T,1024)
  float* out_r = out_s + (size_t)M_TOTAL * DM;      // (B,T,1024)
  float* out_l = out_r + (size_t)M_TOTAL * DM;      // (B,T)

  // Phase 1: parallel pre-GEMM over the r_src part of lat_w1
  pre_kernel<<<M_TOTAL / 64, 128, 0, stream>>>(r_src, lat_w1, lat_b1, pre);

  // Phase 2: sequential scan (only the truly recurrent work)
  scan_kernel<<<1, 256, 0, stream>>>(pre, lat_w1 + (size_t)DM * DMLP, lat_w2,
                                     lat_b2, gru_wi, gru_bi, gru_wh, gru_bh,
                                     feat);

  // Phase 3: parallel fused head MLPs
  head_kernel<<<dim3(M_TOTAL / 64, 3), 128, 0, stream>>>(
      feat, steer_w1, steer_b1, steer_w2, steer_b2, rec_w1, rec_b1, rec_w2,
      rec_b2, loss_w1, loss_b1, loss_w2, loss_b2, out_s, out_r, out_l);
}